// MCHCGraphSage_56410100465947
// MI455X (gfx1250) — compile-verified
//
#include <hip/hip_runtime.h>
#include <hip/hip_bf16.h>

#define DIM 64
#define WPAD 68   // padded LDS column stride: bank = (4*col + k) % 64 -> conflict-free b64

typedef __attribute__((ext_vector_type(2))) float v2f;
typedef __attribute__((ext_vector_type(8))) float v8f;

// ---------------------------------------------------------------- utilities

__global__ void zero_f32(float* __restrict__ p, int n) {
    int i = blockIdx.x * blockDim.x + threadIdx.x;
    if (i < n) p[i] = 0.0f;
}

__global__ void edge_count(const int* __restrict__ dst, float* __restrict__ cnt, int e) {
    int i = blockIdx.x * blockDim.x + threadIdx.x;
    if (i < e) atomicAdd(&cnt[dst[i]], 1.0f);
}

__global__ void inv_count(const float* __restrict__ cnt, float* __restrict__ inv, int n) {
    int i = blockIdx.x * blockDim.x + threadIdx.x;
    if (i < n) inv[i] = 1.0f / fmaxf(cnt[i], 1.0f);
}

// One wave32 per edge; lane d handles features d and d+32.
// Gathers are 128B-coalesced per half-row; atomics resolve in L2 (12.8MB target).
__global__ void scatter_feat(const float* __restrict__ X, const int* __restrict__ src,
                             const int* __restrict__ dst, float* __restrict__ msg, int e) {
    int t = blockIdx.x * blockDim.x + threadIdx.x;
    int edge = t >> 5;
    int lane = t & 31;
    if (edge >= e) return;
    int s = src[edge];
    int d = dst[edge];
    float v0 = X[(size_t)s * DIM + lane];
    float v1 = X[(size_t)s * DIM + lane + 32];
    atomicAdd(&msg[(size_t)d * DIM + lane], v0);
    atomicAdd(&msg[(size_t)d * DIM + lane + 32], v1);
}

// ------------------------------------------------- fused SAGE layer GEMM
// H[row, :] = ReLU( (msg[row,:] * inv[row]) @ Wl + X[row,:] @ Ws + bl + bs )
// One wave per 16-node strip; 4 accumulator tiles cover the 64 output cols.
// f32 WMMA 16x16x4; weights staged transposed+padded in LDS so each lane's
// B pair (k, k+1 of one column) is one conflict-free ds_load_b64.
__global__ void __launch_bounds__(256)
sage_gemm(const float* __restrict__ M, const float* __restrict__ X,
          const float* __restrict__ inv,
          const float* __restrict__ Wl, const float* __restrict__ bl,
          const float* __restrict__ Ws, const float* __restrict__ bs,
          float* __restrict__ H, int n) {
    __shared__ float sWl[DIM * WPAD];   // sWl[col*WPAD + k] = Wl[k*DIM + col]
    __shared__ float sWs[DIM * WPAD];
    for (int i = threadIdx.x; i < DIM * DIM; i += blockDim.x) {
        int k = i >> 6;
        int c = i & 63;
        sWl[c * WPAD + k] = Wl[i];      // coalesced global read, transposed LDS write
        sWs[c * WPAD + k] = Ws[i];
    }
    __syncthreads();

    int lane   = threadIdx.x & 31;
    int waveId = blockIdx.x * (blockDim.x >> 5) + (threadIdx.x >> 5);
    int row0   = waveId * 16;
    if (row0 >= n) return;          // wave-uniform: EXEC stays all-1s for WMMA

    int hi  = lane >> 4;            // 0: K=k0..k0+1   1: K=k0+2..k0+3  (A/B layout)
    int l16 = lane & 15;

    v8f acc[4];
#pragma unroll
    for (int t = 0; t < 4; ++t) {
        float bias = bl[t * 16 + l16] + bs[t * 16 + l16];
#pragma unroll
        for (int r = 0; r < 8; ++r) acc[t][r] = bias;
    }

    int arow = row0 + l16;
    float sc = inv[arow];
    const float* mrow = M + (size_t)arow * DIM;
    const float* xrow = X + (size_t)arow * DIM;

#pragma unroll
    for (int k0 = 0; k0 < DIM; k0 += 4) {
        int ka = k0 + hi * 2;
        v2f am, ax;
        am.x = mrow[ka] * sc;       // fold mean-normalization into A operand
        am.y = mrow[ka + 1] * sc;
        ax.x = xrow[ka];
        ax.y = xrow[ka + 1];
#pragma unroll
        for (int t = 0; t < 4; ++t) {
            int col = t * 16 + l16;
            v2f bm = *(const v2f*)&sWl[col * WPAD + ka];   // single ds_load_b64
            v2f bx = *(const v2f*)&sWs[col * WPAD + ka];
            acc[t] = __builtin_amdgcn_wmma_f32_16x16x4_f32(
                false, am, false, bm, (short)0, acc[t], false, false);
            acc[t] = __builtin_amdgcn_wmma_f32_16x16x4_f32(
                false, ax, false, bx, (short)0, acc[t], false, false);
        }
    }

    // C/D layout: VGPR r -> row r (lanes 0-15) / row r+8 (lanes 16-31), col = lane%16
#pragma unroll
    for (int t = 0; t < 4; ++t) {
        int col = t * 16 + l16;
#pragma unroll
        for (int r = 0; r < 8; ++r) {
            int row = row0 + r + hi * 8;
            H[(size_t)row * DIM + col] = fmaxf(acc[t][r], 0.0f);
        }
    }
}

// ------------------------------------------------- final 64 -> 1 projection
__global__ void sage_out(const float* __restrict__ M, const float* __restrict__ X,
                         const float* __restrict__ inv,
                         const float* __restrict__ Wl, const float* __restrict__ bl,
                         const float* __restrict__ Ws, const float* __restrict__ bs,
                         float* __restrict__ out, int n) {
    int t = blockIdx.x * blockDim.x + threadIdx.x;
    int node = t >> 5;
    int lane = t & 31;
    if (node >= n) return;
    float sc = inv[node];
    const float* mrow = M + (size_t)node * DIM;
    const float* xrow = X + (size_t)node * DIM;
    float p = mrow[lane] * sc * Wl[lane] + xrow[lane] * Ws[lane]
            + mrow[lane + 32] * sc * Wl[lane + 32] + xrow[lane + 32] * Ws[lane + 32];
#pragma unroll
    for (int off = 16; off > 0; off >>= 1)
        p += __shfl_xor(p, off, 32);
    if (lane == 0) out[node] = fmaxf(p + bl[0] + bs[0], 0.0f);
}

// ---------------------------------------------------------------- launcher

extern "C" void kernel_launch(void* const* d_in, const int* in_sizes, int n_in,
                              void* d_out, int out_size, void* d_ws, size_t ws_size,
                              hipStream_t stream) {
    (void)n_in; (void)out_size; (void)ws_size;
    const float* x   = (const float*)d_in[0];
    const float* Wl0 = (const float*)d_in[1];
    const float* bl0 = (const float*)d_in[2];
    const float* Ws0 = (const float*)d_in[3];
    const float* bs0 = (const float*)d_in[4];
    const float* Wl1 = (const float*)d_in[5];
    const float* bl1 = (const float*)d_in[6];
    const float* Ws1 = (const float*)d_in[7];
    const float* bs1 = (const float*)d_in[8];
    const float* Wl2 = (const float*)d_in[9];
    const float* bl2 = (const float*)d_in[10];
    const float* Ws2 = (const float*)d_in[11];
    const float* bs2 = (const float*)d_in[12];
    const int*   ei  = (const int*)d_in[13];   // edge_index flattened [2, E]
    // d_in[14] = class_edge_index: unused by the reference forward.

    int n = in_sizes[0] / DIM;      // 50000
    int e = in_sizes[13] / 2;       // 800000
    const int* src = ei;
    const int* dst = ei + e;

    float* msg = (float*)d_ws;                    // [n, 64]
    float* h0  = msg + (size_t)n * DIM;           // [n, 64]
    float* h1  = h0 + (size_t)n * DIM;            // [n, 64]
    float* cnt = h1 + (size_t)n * DIM;            // [n]
    float* inv = cnt + n;                         // [n]
    float* out = (float*)d_out;                   // [n]

    auto cdiv = [](int a, int b) { return (a + b - 1) / b; };
    dim3 blk(256);
    int gN    = cdiv(n, 256);
    int gE    = cdiv(e, 256);
    int gScat = cdiv(e * 32, 256);                // wave per edge
    int gGemm = cdiv(cdiv(n, 16), 8);             // wave per 16-row strip, 8 waves/block
    int gOut  = cdiv(n * 32, 256);                // wave per node

    // degree counts (same edges for all layers)
    zero_f32  <<<gN, blk, 0, stream>>>(cnt, n);
    edge_count<<<gE, blk, 0, stream>>>(dst, cnt, e);
    inv_count <<<gN, blk, 0, stream>>>(cnt, inv, n);

    // layer 0
    zero_f32    <<<cdiv(n * DIM, 256), blk, 0, stream>>>(msg, n * DIM);
    scatter_feat<<<gScat, blk, 0, stream>>>(x, src, dst, msg, e);
    sage_gemm   <<<gGemm, blk, 0, stream>>>(msg, x, inv, Wl0, bl0, Ws0, bs0, h0, n);

    // layer 1
    zero_f32    <<<cdiv(n * DIM, 256), blk, 0, stream>>>(msg, n * DIM);
    scatter_feat<<<gScat, blk, 0, stream>>>(h0, src, dst, msg, e);
    sage_gemm   <<<gGemm, blk, 0, stream>>>(msg, h0, inv, Wl1, bl1, Ws1, bs1, h1, n);

    // layer 2 (64 -> 1)
    zero_f32    <<<cdiv(n * DIM, 256), blk, 0, stream>>>(msg, n * DIM);
    scatter_feat<<<gScat, blk, 0, stream>>>(h1, src, dst, msg, e);
    sage_out    <<<gOut, blk, 0, stream>>>(msg, h1, inv, Wl2, bl2, Ws2, bs2, out, n);
}